// GCN_86535001079836
// MI455X (gfx1250) — compile-verified
//
#include <hip/hip_runtime.h>
#include <math.h>

typedef __attribute__((ext_vector_type(2))) float v2f;
typedef __attribute__((ext_vector_type(8))) float v8f;

#define NFEAT 512
#define NHID  128
#define NCLS  40
#define NCLSP 48   // padded to 3 WMMA N-tiles

// ---------------------------------------------------------------------------
// small utility kernels
// ---------------------------------------------------------------------------
__global__ void zero_f32(float* __restrict__ p, int count) {
    int t = blockIdx.x * 256 + threadIdx.x;
    if (t < count) p[t] = 0.0f;
}

__global__ void deg_init(float* __restrict__ deg, int n) {
    int t = blockIdx.x * 256 + threadIdx.x;
    if (t < n) deg[t] = 1.0f;            // self loop
}

__global__ void deg_edges(const int* __restrict__ dst, float* __restrict__ deg, int nE) {
    int t = blockIdx.x * 256 + threadIdx.x;
    if (t < nE) atomicAdd(&deg[dst[t]], 1.0f);
}

__global__ void deg_rsqrt(float* __restrict__ deg, int n) {
    int t = blockIdx.x * 256 + threadIdx.x;
    if (t < n) deg[t] = rsqrtf(deg[t]);  // deg >= 1 always
}

// ---------------------------------------------------------------------------
// GEMM1: h[n,128] = (x .* mask)[n,512] @ W1[512,128]   (fp32 WMMA 16x16x4)
// sX : [128 rows][64 k]  stride 68 -> conflict-free ds_load_b64 A-fragments
// sWt: [128 cols][64 k]  stride 68 -> B stored K-contiguous per column, so a
//      B-fragment is a single ds_load_b64 (no cross-row packing movs)
// ---------------------------------------------------------------------------
#define SXLD 68
#define SWTLD 68

__global__ __launch_bounds__(256) void gemm1_wmma(
        const float* __restrict__ x, const float* __restrict__ mask,
        const float* __restrict__ W1, float* __restrict__ h, int n) {
    __shared__ float sX[128 * SXLD];
    __shared__ float sWt[128 * SWTLD];
    const int tid  = threadIdx.x;
    const int lane = tid & 31;
    const int wave = tid >> 5;
    const int m    = lane & 15;     // M (A) / N (B) index within tile
    const int kg   = lane >> 4;     // K-group {0,1}
    const int row0 = blockIdx.x * 128;

    const v8f vzero = {0.f,0.f,0.f,0.f,0.f,0.f,0.f,0.f};
    v8f acc[8];
#pragma unroll
    for (int nt = 0; nt < 8; ++nt) acc[nt] = vzero;

    for (int kc = 0; kc < NFEAT; kc += 64) {
        // stage x*mask tile [128 x 64] -> LDS (8 float4 per thread)
#pragma unroll
        for (int it = 0; it < 8; ++it) {
            int idx = (it * 256 + tid) * 4;          // 0..8191
            int r = idx >> 6, c = idx & 63;
            int gr = row0 + r; if (gr >= n) gr = n - 1;
            const float4 xv = *(const float4*)&x[(size_t)gr * NFEAT + kc + c];
            const float4 mv = *(const float4*)&mask[(size_t)gr * NFEAT + kc + c];
            float4 v; v.x = xv.x*mv.x; v.y = xv.y*mv.y; v.z = xv.z*mv.z; v.w = xv.w*mv.w;
            *(float4*)&sX[r * SXLD + c] = v;
        }
        // stage W1 chunk [64 x 128] transposed -> sWt[col][k]
#pragma unroll
        for (int it = 0; it < 8; ++it) {
            int idx = (it * 256 + tid) * 4;          // 0..8191
            int r = idx >> 7, c0 = idx & 127;        // r = k-in-chunk, c0 = col
            const float4 wv = *(const float4*)&W1[(size_t)(kc + r) * NHID + c0];
            sWt[(c0 + 0) * SWTLD + r] = wv.x;
            sWt[(c0 + 1) * SWTLD + r] = wv.y;
            sWt[(c0 + 2) * SWTLD + r] = wv.z;
            sWt[(c0 + 3) * SWTLD + r] = wv.w;
        }
        __syncthreads();

        for (int k = 0; k < 64; k += 4) {
            const int krow = k + 2 * kg;
            // A 16x4 fp32 fragment: lane = M + 16*kgroup, holds K=2*kg, 2*kg+1
            const v2f a = *(const v2f*)&sX[(wave * 16 + m) * SXLD + krow];
#pragma unroll
            for (int nt = 0; nt < 8; ++nt) {
                // B 4x16 fp32 fragment: lane = N + 16*kgroup, K-contiguous
                const v2f b = *(const v2f*)&sWt[(nt * 16 + m) * SWTLD + krow];
                acc[nt] = __builtin_amdgcn_wmma_f32_16x16x4_f32(
                    false, a, false, b, (short)0, acc[nt], false, false);
            }
        }
        __syncthreads();
    }
    // C/D layout: VGPR j holds row M = j + 8*kg, col N = lane&15
    const int rbase = row0 + wave * 16 + 8 * kg;
#pragma unroll
    for (int nt = 0; nt < 8; ++nt)
#pragma unroll
        for (int j = 0; j < 8; ++j) {
            int r = rbase + j;
            if (r < n) h[(size_t)r * NHID + nt * 16 + m] = acc[nt][j];
        }
}

// ---------------------------------------------------------------------------
// GEMM2: h2[n,40] = relu1[n,128] @ W2[128,40]   (N padded to 48)
// X tile staged with CDNA5 async global->LDS DMA (ASYNCcnt / s_wait_asynccnt)
// ---------------------------------------------------------------------------
#define SX2LD 132  // 128 + 4 pad ; 132*4 bytes is 16B-multiple -> b128 ok
#define SW2LD 132  // transposed W2: [48 cols][128 k + 4 pad]

__global__ __launch_bounds__(256) void gemm2_wmma(
        const float* __restrict__ a_in, const float* __restrict__ W2,
        float* __restrict__ h2, int n) {
    __shared__ float sX[128 * SX2LD];
    __shared__ float sWt[NCLSP * SW2LD];
    const int tid  = threadIdx.x;
    const int lane = tid & 31;
    const int wave = tid >> 5;
    const int m    = lane & 15;
    const int kg   = lane >> 4;
    const int row0 = blockIdx.x * 128;

    // stage relu1 tile [128 x 128] via async 16B global->LDS copies
#pragma unroll
    for (int it = 0; it < 16; ++it) {
        int idx = (it * 256 + tid) * 4;
        int r = idx >> 7, c = idx & 127;
        int gr = row0 + r; if (gr >= n) gr = n - 1;
        const float* gp = &a_in[(size_t)gr * NHID + c];
        unsigned lds_off = (unsigned)(size_t)(&sX[r * SX2LD + c]);
        asm volatile("global_load_async_to_lds_b128 %0, %1, off"
                     :: "v"(lds_off), "v"(gp) : "memory");
    }
    // stage W2 [128 x 40] transposed + zero-padded -> sWt[col][k]
#pragma unroll
    for (int it = 0; it < 24; ++it) {
        int idx = it * 256 + tid;                 // 0..6143
        int c = idx >> 7, r = idx & 127;          // c = col (0..47), r = k
        sWt[c * SW2LD + r] = (c < NCLS) ? W2[r * NCLS + c] : 0.0f;
    }
    asm volatile("s_wait_asynccnt 0x0" ::: "memory");
    __syncthreads();

    const v8f vzero = {0.f,0.f,0.f,0.f,0.f,0.f,0.f,0.f};
    v8f acc[3];
#pragma unroll
    for (int nt = 0; nt < 3; ++nt) acc[nt] = vzero;

    for (int k = 0; k < NHID; k += 4) {
        const int krow = k + 2 * kg;
        const v2f a = *(const v2f*)&sX[(wave * 16 + m) * SX2LD + krow];
#pragma unroll
        for (int nt = 0; nt < 3; ++nt) {
            const v2f b = *(const v2f*)&sWt[(nt * 16 + m) * SW2LD + krow];
            acc[nt] = __builtin_amdgcn_wmma_f32_16x16x4_f32(
                false, a, false, b, (short)0, acc[nt], false, false);
        }
    }
    const int rbase = row0 + wave * 16 + 8 * kg;
#pragma unroll
    for (int nt = 0; nt < 3; ++nt)
#pragma unroll
        for (int j = 0; j < 8; ++j) {
            int r = rbase + j, c = nt * 16 + m;
            if (r < n && c < NCLS) h2[(size_t)r * NCLS + c] = acc[nt][j];
        }
}

// ---------------------------------------------------------------------------
// edge aggregation (scatter-add) — L2-resident random traffic
// ---------------------------------------------------------------------------
__global__ void agg1_edges(const float* __restrict__ h, const float* __restrict__ dis,
                           const int* __restrict__ src, const int* __restrict__ dst,
                           float* __restrict__ agg, int nE) {
    int t = blockIdx.x * 256 + threadIdx.x;     // one wave per edge, lane -> 4 cols
    int e = t >> 5;
    if (e >= nE) return;
    int c4 = (t & 31) * 4;
    int s = src[e], d = dst[e];
    float norm = dis[s] * dis[d];
    const float4 hv = *(const float4*)&h[(size_t)s * NHID + c4];
    float* o = &agg[(size_t)d * NHID + c4];
    atomicAdd(o + 0, hv.x * norm);
    atomicAdd(o + 1, hv.y * norm);
    atomicAdd(o + 2, hv.z * norm);
    atomicAdd(o + 3, hv.w * norm);
}

__global__ void agg2_edges(const float* __restrict__ h2, const float* __restrict__ dis,
                           const int* __restrict__ src, const int* __restrict__ dst,
                           float* __restrict__ out, int nE) {
    int t = blockIdx.x * 256 + threadIdx.x;     // thread per (edge, class)
    int e = t / NCLS;
    if (e >= nE) return;
    int c = t - e * NCLS;
    int s = src[e], d = dst[e];
    float norm = dis[s] * dis[d];
    atomicAdd(&out[(size_t)d * NCLS + c], h2[(size_t)s * NCLS + c] * norm);
}

// self-loop term + bias (+ ReLU for layer 1), in place on agg buffer
__global__ void epi1_relu(float* __restrict__ agg, const float* __restrict__ h,
                          const float* __restrict__ dis, const float* __restrict__ b1, int n) {
    int t = blockIdx.x * 256 + threadIdx.x;
    if (t >= n * NHID) return;
    int i = t >> 7, c = t & 127;
    float di = dis[i];
    float v = agg[t] + h[t] * di * di + b1[c];
    agg[t] = v > 0.0f ? v : 0.0f;
}

__global__ void epi2_bias(float* __restrict__ out, const float* __restrict__ h2,
                          const float* __restrict__ dis, const float* __restrict__ b2, int n) {
    int t = blockIdx.x * 256 + threadIdx.x;
    if (t >= n * NCLS) return;
    int i = t / NCLS, c = t - i * NCLS;
    float di = dis[i];
    out[t] += h2[t] * di * di + b2[c];
}

// ---------------------------------------------------------------------------
// launch
// ---------------------------------------------------------------------------
extern "C" void kernel_launch(void* const* d_in, const int* in_sizes, int n_in,
                              void* d_out, int out_size, void* d_ws, size_t ws_size,
                              hipStream_t stream) {
    const float* x    = (const float*)d_in[0];
    const float* mask = (const float*)d_in[1];
    const float* W1   = (const float*)d_in[2];
    const float* b1   = (const float*)d_in[3];
    const float* W2   = (const float*)d_in[4];
    const float* b2   = (const float*)d_in[5];
    const int*   ei   = (const int*)d_in[6];

    const int n  = in_sizes[0] / NFEAT;     // 100000
    const int nE = in_sizes[6] / 2;         // 1600000
    const int* src = ei;
    const int* dst = ei + nE;

    float* ws   = (float*)d_ws;
    float* dis  = ws;                       // n floats (deg -> rsqrt in place)
    float* h1   = ws + 100352;              // n*128
    float* agg1 = h1 + (size_t)n * NHID;    // n*128 (becomes relu1 in place)
    float* h2   = h1;                       // h1 dead after epi1; reuse
    float* out  = (float*)d_out;            // n*40 fp32

    const int B = 256;
    const int mblocks = (n + 127) / 128;

    // degrees -> dis = deg^-1/2
    hipLaunchKernelGGL(deg_init,  dim3((n + B - 1) / B), dim3(B), 0, stream, dis, n);
    hipLaunchKernelGGL(deg_edges, dim3((nE + B - 1) / B), dim3(B), 0, stream, dst, dis, nE);
    hipLaunchKernelGGL(deg_rsqrt, dim3((n + B - 1) / B), dim3(B), 0, stream, dis, n);

    // zero accumulation buffers
    hipLaunchKernelGGL(zero_f32, dim3((n * NHID + B - 1) / B), dim3(B), 0, stream, agg1, n * NHID);
    hipLaunchKernelGGL(zero_f32, dim3((n * NCLS + B - 1) / B), dim3(B), 0, stream, out, n * NCLS);

    // layer 1
    hipLaunchKernelGGL(gemm1_wmma, dim3(mblocks), dim3(B), 0, stream, x, mask, W1, h1, n);
    hipLaunchKernelGGL(agg1_edges, dim3((nE * 32 + B - 1) / B), dim3(B), 0, stream,
                       h1, dis, src, dst, agg1, nE);
    hipLaunchKernelGGL(epi1_relu, dim3((n * NHID + B - 1) / B), dim3(B), 0, stream,
                       agg1, h1, dis, b1, n);

    // layer 2
    hipLaunchKernelGGL(gemm2_wmma, dim3(mblocks), dim3(B), 0, stream, agg1, W2, h2, n);
    hipLaunchKernelGGL(agg2_edges, dim3((nE * NCLS + B - 1) / B), dim3(B), 0, stream,
                       h2, dis, src, dst, out, nE);
    hipLaunchKernelGGL(epi2_bias, dim3((n * NCLS + B - 1) / B), dim3(B), 0, stream,
                       out, h2, dis, b2, n);
}